// S4DKernel_54485955117196
// MI455X (gfx1250) — compile-verified
//
#include <hip/hip_runtime.h>
#include <math.h>

typedef __attribute__((ext_vector_type(2))) float v2f;
typedef __attribute__((ext_vector_type(4))) float v4f;
typedef __attribute__((ext_vector_type(8))) float v8f;

#define HD 1024   // D_MODEL
#define NS 32     // d_state/2
#define LL 4096   // sequence length

// One block (4 wave32) per h.  K[h, 16a+b] = Re( A(16x32) * B(32x256) ) with
//   A[b,n] = 2*Ceff[h,n]*exp(dtA*b)   (2x and Im-sign folded in)
//   B[n,a] = exp(dtA*16a)             (pure geometric recurrences)
// Each wave owns 4 consecutive column-tiles; 16 chained V_WMMA_F32_16X16X4_F32
// per 16x16 tile (K=32, complex => 2 real GEMMs).  All accurate-range-reduction
// sincosf calls are hoisted out of the tile loop.
__global__ __launch_bounds__(128)
void s4d_wmma_kernel(const float* __restrict__ log_dt,
                     const float* __restrict__ C_ri,
                     const float* __restrict__ log_A_real,
                     const float* __restrict__ A_imag,
                     float* __restrict__ out)
{
    __shared__ float ArS[16][33];        // 2*Re(Ceff * exp(dtA*b))
    __shared__ float AiS[16][33];        // -2*Im(Ceff * exp(dtA*b))
    __shared__ float BrS[4][32][17];     // per-wave B tiles: Re(exp(dtA*16a))
    __shared__ float BiS[4][32][17];     // per-wave B tiles: Im

    const int h    = blockIdx.x;
    const int tid  = threadIdx.x;
    const int lane = tid & 31;           // wave lane; == n during setup
    const int wv   = tid >> 5;           // wave id 0..3

    // ---- per-(h,n) setup (replicated across the 4 waves) ----
    const float dt = __expf(log_dt[h]);
    const float ar = -__expf(log_A_real[h * NS + lane]);  // Re(A) < 0
    const float ai = A_imag[h * NS + lane];               // Im(A)
    const float x  = dt * ar;                             // Re(dtA)
    const float y  = dt * ai;                             // Im(dtA)
    const float cr = C_ri[(h * NS + lane) * 2 + 0];
    const float ci = C_ri[(h * NS + lane) * 2 + 1];

    // w = exp(dtA) - 1 ;  g = w / A ;  Ceff = C * g  (scaled by 2)
    float e1 = __expf(x);
    float s1, c1; sincosf(y, &s1, &c1);
    const float wr  = e1 * c1 - 1.0f;
    const float wi  = e1 * s1;
    const float inv = 1.0f / (ar * ar + ai * ai);
    const float gr  = (wr * ar + wi * ai) * inv;
    const float gi  = (wi * ar - wr * ai) * inv;
    const float fr  = 2.0f * (cr * gr - ci * gi);
    const float fi  = 2.0f * (cr * gi + ci * gr);

    // ---- A rows: each wave fills 4 of the 16 rows ----
    for (int i = 0; i < 4; ++i) {
        int b = wv * 4 + i;
        float eb = __expf(x * (float)b);
        float sb, cb; sincosf(y * (float)b, &sb, &cb);
        float qr = eb * cb, qi = eb * sb;
        ArS[b][lane] = fr * qr - fi * qi;
        AiS[b][lane] = -(fr * qi + fi * qr);   // pre-negated imaginary part
    }
    __syncthreads();

    // ---- A fragments (shared by all 4 waves, constant over tiles) ----
    // f32 16x4 A layout: lane%16 = row M, VGPR pair = {K=k0,K=k0+1},
    // lanes 16..31 carry K offset +2.
    const int arow   = lane & 15;
    const int k0base = (lane >> 4) * 2;
    v2f afr[8], afi[8];
    for (int kk = 0; kk < 8; ++kk) {
        int k0 = 4 * kk + k0base;
        afr[kk] = v2f{ ArS[arow][k0], ArS[arow][k0 + 1] };
        afi[kk] = v2f{ AiS[arow][k0], AiS[arow][k0 + 1] };
    }

    // ---- geometric steps (accurate anchors, computed once per thread) ----
    // within-tile step  s    = exp(16*dtA)
    float es = __expf(16.0f * x);
    float ss, cs; sincosf(16.0f * y, &ss, &cs);
    const float sr = es * cs, si = es * ss;
    // tile-to-tile step S256 = exp(256*dtA)
    float eS = __expf(256.0f * x);
    float sS, cS; sincosf(256.0f * y, &sS, &cS);
    const float Sr = eS * cS, Si = eS * sS;
    // initial anchor for this wave's first tile t0 = 4*wv: exp(dtA*256*t0)
    float fa = (float)(1024 * wv);
    float pm = __expf(x * fa);
    float sa, ca; sincosf(y * fa, &sa, &ca);
    float qr0 = pm * ca, qi0 = pm * sa;

    const int col = lane & 15;           // B/D column N
    const int r0  = (lane >> 4) * 8;     // D rows: M = r0..r0+7
    float* const outp = out + (size_t)h * LL;

    for (int tt = 0; tt < 4; ++tt) {
        const int t = wv * 4 + tt;       // global tile index 0..15
        __syncthreads();                 // prior tile's B reads complete

        // ---- B tile: branchless geometric recurrence from anchor ----
        float pr = qr0, pi = qi0;
        for (int j = 0; j < 16; ++j) {
            BrS[wv][lane][j] = pr;
            BiS[wv][lane][j] = pi;
            float nr = pr * sr - pi * si;          // P *= s
            pi       = pr * si + pi * sr;
            pr       = nr;
        }
        // advance anchor to next tile: q *= S256
        float nq = qr0 * Sr - qi0 * Si;
        qi0      = qr0 * Si + qi0 * Sr;
        qr0      = nq;
        __syncthreads();

        // ---- 16x16x32 complex-real GEMM: 16 chained f32 WMMAs ----
        v8f acc = {};
        for (int kk = 0; kk < 8; ++kk) {
            int k0 = 4 * kk + k0base;
            v2f bfr = v2f{ BrS[wv][k0][col], BrS[wv][k0 + 1][col] };
            v2f bfi = v2f{ BiS[wv][k0][col], BiS[wv][k0 + 1][col] };
            acc = __builtin_amdgcn_wmma_f32_16x16x4_f32(
                false, afr[kk], false, bfr, (short)0, acc, false, false);
            acc = __builtin_amdgcn_wmma_f32_16x16x4_f32(
                false, afi[kk], false, bfi, (short)0, acc, false, false);
        }

        // ---- store: l = 256*t + 16*col + M; each lane owns 8 consecutive l ----
        float* p = outp + 256 * t + 16 * col + r0;   // 32B-aligned
        *(v4f*)(p)     = v4f{ acc[0], acc[1], acc[2], acc[3] };
        *(v4f*)(p + 4) = v4f{ acc[4], acc[5], acc[6], acc[7] };
    }
}

extern "C" void kernel_launch(void* const* d_in, const int* in_sizes, int n_in,
                              void* d_out, int out_size, void* d_ws, size_t ws_size,
                              hipStream_t stream) {
    (void)in_sizes; (void)n_in; (void)out_size; (void)d_ws; (void)ws_size;
    // d_in[0] = L (int scalar, == 4096, unused)
    const float* log_dt     = (const float*)d_in[1];
    const float* C_ri       = (const float*)d_in[2];
    const float* log_A_real = (const float*)d_in[3];
    const float* A_imag     = (const float*)d_in[4];
    float* out = (float*)d_out;

    s4d_wmma_kernel<<<dim3(HD), dim3(128), 0, stream>>>(
        log_dt, C_ri, log_A_real, A_imag, out);
}